// ContrastiveLoss_22333829940001
// MI455X (gfx1250) — compile-verified
//
#include <hip/hip_runtime.h>
#include <stdint.h>

// ---------------------------------------------------------------------------
// CPC contrastive loss on MI455X (gfx1250).
//   z           : (B=8, C=512, T=2048) f32
//   c           : unused by reference
//   predictions : (K=12, B, C, T) f32
//   neg_indices : (K, B*T, 10) i32
// Strategy (memory-bound, ~500MB HBM -> ~21us floor @23.3TB/s):
//   1) transpose z -> zT (B*T, C) row-contiguous in workspace (gathers become
//      coalesced b128 loads; zT (32MB) stays resident in the 192MB L2)
//   2) per (k,b,32-l tile) workgroup: TDM tensor_load_to_lds stages the
//      strided 512x32 pred & z tiles into LDS (HW pad -> 33-dword rows,
//      bank-conflict free), s_wait_tensorcnt + barrier
//   3) positive term via v_wmma_f32_16x16x4_f32 (diagonal of P^T x Z, fp32)
//   4) negative gathers: fp32 VALU dots, wave32 shfl_xor reduction
//   5) deterministic per-block partials -> fixed-order finalize (no f32 atomics)
// ---------------------------------------------------------------------------

#define Bv 8
#define Cv 512
#define Tv 2048
#define Kv 12
#define NN 10
#define LT 32
#define LP 33
#define LTILES (Tv / LT)

typedef float v2f __attribute__((ext_vector_type(2)));
typedef float v8f __attribute__((ext_vector_type(8)));
typedef unsigned int u32x4 __attribute__((ext_vector_type(4)));
typedef int i32x4 __attribute__((ext_vector_type(4)));
typedef int i32x8 __attribute__((ext_vector_type(8)));

// Device-pass-only feature detection with loud (stderr) fallback markers so the
// compile-only loop can confirm which paths lowered.
#if defined(__HIP_DEVICE_COMPILE__)
  #if __has_builtin(__builtin_amdgcn_tensor_load_to_lds)
    #define HAS_TDM 1
  #else
    #define HAS_TDM 0
    #warning "CDNA5 probe: __builtin_amdgcn_tensor_load_to_lds NOT available in device pass; using VALU fallback"
  #endif
  #if __has_builtin(__builtin_amdgcn_wmma_f32_16x16x4_f32)
    #define HAS_WMMA 1
  #else
    #define HAS_WMMA 0
    #warning "CDNA5 probe: __builtin_amdgcn_wmma_f32_16x16x4_f32 NOT available in device pass; using VALU fallback"
  #endif
#else
  #define HAS_TDM 0
  #define HAS_WMMA 0
#endif

__device__ __forceinline__ float softplusf(float x) {
  // -log_sigmoid(-x) = softplus(x), numerically stable
  return fmaxf(x, 0.f) + log1pf(expf(-fabsf(x)));
}

#if HAS_TDM
// 2D TDM descriptor: tile = LT x Cv f32 elements out of a (tensor_d0 x Cv)
// tensor whose dim-0 stride is Tv elements. LDS pad: +1 dword per 32 dwords
// => LDS row stride LP=33 dwords (matches tile[Cv][LP] layout, no bank conflicts).
__device__ __forceinline__ void tdm_load_tile_2d(unsigned lds_off, const float* gptr,
                                                 int tensor_d0) {
  unsigned long long ga = (unsigned long long)(uintptr_t)gptr;
  u32x4 g0;
  g0[0] = 1u;                                                   // count=1 valid descriptor
  g0[1] = lds_off;                                              // lds_addr (bytes)
  g0[2] = (unsigned)ga;                                         // global_addr[31:0]
  g0[3] = ((unsigned)(ga >> 32) & 0x1FFFFFFu) | 0x80000000u;    // addr[56:32] | type=2
  i32x8 g1;
  g1[0] = (2 << 16) | (1 << 20) | (4 << 22);  // data_size=4B, pad_enable, every 32dw, +1dw
  g1[1] = (tensor_d0 & 0xFFFF) << 16;                           // tensor_dim0[15:0]
  g1[2] = ((tensor_d0 >> 16) & 0xFFFF) | (Cv << 16);            // dim0[31:16] | dim1[15:0]
  g1[3] = (LT << 16);                                           // dim1[31:16]=0 | tile_dim0
  g1[4] = Cv;                                                   // tile_dim1 | tile_dim2=0
  g1[5] = Tv;                                                   // tensor_dim0_stride[31:0]
  g1[6] = 0;                                                    // stride0[47:32] | stride1 lo
  g1[7] = 0;
  i32x4 gz = {};
#if __clang_major__ >= 23
  i32x8 gz8 = {};
  __builtin_amdgcn_tensor_load_to_lds(g0, g1, gz, gz, gz8, 0);
#else
  __builtin_amdgcn_tensor_load_to_lds(g0, g1, gz, gz, 0);
#endif
}
#endif

// ---------------------------------------------------------------------------
// Kernel 1: transpose z (B,C,T) -> zT (B*T, C) row-contiguous
// ---------------------------------------------------------------------------
__global__ __launch_bounds__(256) void cpc_transpose_z(const float* __restrict__ z,
                                                       float* __restrict__ zT) {
  __shared__ float tile[32][33];
  const int b = blockIdx.z;
  const int t0 = blockIdx.x * 32, c0 = blockIdx.y * 32;
  const int tx = threadIdx.x, ty = threadIdx.y;  // 32 x 8
  const float* zb = z + (size_t)b * Cv * Tv;
  float* zTb = zT + (size_t)b * Tv * Cv;
#pragma unroll
  for (int i = 0; i < 32; i += 8)
    tile[ty + i][tx] = zb[(size_t)(c0 + ty + i) * Tv + (t0 + tx)];
  __syncthreads();
#pragma unroll
  for (int i = 0; i < 32; i += 8)
    zTb[(size_t)(t0 + ty + i) * Cv + (c0 + tx)] = tile[tx][ty + i];
}

// ---------------------------------------------------------------------------
// Kernel 2: main loss kernel. grid = (LTILES, B, K), block = 256 (8 waves)
// ---------------------------------------------------------------------------
__global__ __launch_bounds__(256) void cpc_main(const float* __restrict__ z,
                                                const float* __restrict__ pred,
                                                const int* __restrict__ negidx,
                                                const float* __restrict__ zT,
                                                float* __restrict__ accum) {
  __shared__ float tileP[Cv][LP];   // pred[k-1,b,c,l0+lt]
  __shared__ float tileZ[Cv][LP];   // z[b,c,l0+k+lt]
  __shared__ float sPosArr[LT];
  __shared__ float sNegArr[8];

  const int kk = blockIdx.z;            // k = kk+1
  const int b = blockIdx.y;
  const int l0 = blockIdx.x * LT;
  const int k = kk + 1;
  const int Lk = Tv - k;
  const int tid = threadIdx.x;
  const int wid = tid >> 5, lane = tid & 31;

  float* slot = accum + (((size_t)kk * Bv + b) * LTILES + blockIdx.x) * 2;
  if (l0 >= Lk) {  // (dead for these shapes, kept for safety; block-uniform)
    if (tid == 0) { slot[0] = 0.f; slot[1] = 0.f; }
    return;
  }

  const float* predBase = pred + ((size_t)kk * Bv + b) * (size_t)Cv * Tv;
  const float* zBase = z + (size_t)b * (size_t)Cv * Tv;

#if HAS_TDM
  if (tid < 32) {  // wave 0 issues both DMA tiles; EXEC ignored by TDM
    tdm_load_tile_2d((unsigned)(uintptr_t)&tileP[0][0], predBase + l0, Tv - l0);
    tdm_load_tile_2d((unsigned)(uintptr_t)&tileZ[0][0], zBase + (l0 + k), Tv - (l0 + k));
    __builtin_amdgcn_s_wait_tensorcnt(0);
  }
#else
  for (int i = tid; i < Cv * LT; i += 256) {
    const int cc = i >> 5, lt = i & 31;
    tileP[cc][lt] = predBase[(size_t)cc * Tv + (l0 + lt)];
    const int lz = l0 + k + lt;
    tileZ[cc][lt] = (lz < Tv) ? zBase[(size_t)cc * Tv + lz] : 0.f;
  }
#endif
  __syncthreads();

  // ---- positive term: diag of P^T x Z via v_wmma_f32_16x16x4_f32 ----------
#if HAS_WMMA
  if (wid < 2) {  // wave-uniform branch; EXEC all ones inside
    const int lt0 = wid * 16;
    const int half = (lane & 16) ? 2 : 0;     // A/B 16x4 f32 operand layout
    const int col = lt0 + (lane & 15);
    v8f acc = {};
    for (int c0 = 0; c0 < Cv; c0 += 4) {
      v2f a, bb;
      a[0] = tileP[c0 + half][col];
      a[1] = tileP[c0 + half + 1][col];
      bb[0] = tileZ[c0 + half][col];
      bb[1] = tileZ[c0 + half + 1][col];
      acc = __builtin_amdgcn_wmma_f32_16x16x4_f32(false, a, false, bb,
                                                  (short)0, acc, false, false);
    }
    // diagonal D[m,m]: m<8 -> lane m, vgpr m; m>=8 -> lane m+8, vgpr m-8
    int m = -1;
    if (lane < 8) m = lane;
    else if (lane >= 16 && lane < 24) m = lane - 8;
    if (m >= 0) {
      const float d = acc[(lane < 8) ? lane : (lane - 16)];
      const int lt = lt0 + m;
      const int l = l0 + lt;
      sPosArr[lt] = (l < Lk) ? softplusf(-d) : 0.f;  // -log_sigmoid(pos_sim)
    }
  }
#else
  if (tid < 32) {
    const int lt = lane, l = l0 + lt;
    if (l < Lk) {
      float d = 0.f;
      for (int cc = 0; cc < Cv; ++cc) d += tileP[cc][lt] * tileZ[cc][lt];
      sPosArr[lt] = softplusf(-d);
    } else {
      sPosArr[lt] = 0.f;
    }
  }
#endif

  // ---- negative term: 10 gathered fp32 dots per row, wave32 reduction -----
  float wsum = 0.f;
  for (int r = 0; r < 4; ++r) {
    const int lt = wid * 4 + r;     // wave-uniform
    const int l = l0 + lt;
    if (l >= Lk) continue;
    float preg[16];
#pragma unroll
    for (int q = 0; q < 16; ++q) preg[q] = tileP[lane * 16 + q][lt];
    const size_t nrow = (size_t)b * Lk + l;
    const int* idxp = negidx + ((size_t)kk * (Bv * Tv) + nrow) * NN;
#pragma unroll
    for (int j = 0; j < NN; ++j) {
      const int rr = idxp[j];
      const float4* zr = (const float4*)(zT + (size_t)rr * Cv) + lane * 4;
      float acc = 0.f;
#pragma unroll
      for (int q = 0; q < 4; ++q) {
        const float4 v = zr[q];
        acc += v.x * preg[q * 4 + 0] + v.y * preg[q * 4 + 1] +
               v.z * preg[q * 4 + 2] + v.w * preg[q * 4 + 3];
      }
      for (int off = 16; off; off >>= 1) acc += __shfl_xor(acc, off, 32);
      if (lane == 0) wsum += softplusf(acc);  // -log_sigmoid(-neg_sim)
    }
  }
  if (lane == 0) sNegArr[wid] = wsum;

  __syncthreads();
  if (wid == 0) {  // deterministic fixed-tree block reduction
    float ps = sPosArr[lane];
    float ns = (lane < 8) ? sNegArr[lane] : 0.f;
    for (int off = 16; off; off >>= 1) {
      ps += __shfl_xor(ps, off, 32);
      ns += __shfl_xor(ns, off, 32);
    }
    if (lane == 0) { slot[0] = ps; slot[1] = ns; }
  }
}

// ---------------------------------------------------------------------------
// Kernel 3: deterministic finalize. 12 waves, one per k.
// ---------------------------------------------------------------------------
__global__ __launch_bounds__(384) void cpc_finalize(const float* __restrict__ accum,
                                                    float* __restrict__ out) {
  __shared__ float sk[Kv];
  const int wid = threadIdx.x >> 5, lane = threadIdx.x & 31;
  if (wid < Kv) {
    const float* base = accum + (size_t)wid * (Bv * LTILES) * 2;
    float ps = 0.f, ns = 0.f;
    for (int i = lane; i < Bv * LTILES; i += 32) {
      ps += base[2 * i];
      ns += base[2 * i + 1];
    }
    for (int off = 16; off; off >>= 1) {
      ps += __shfl_xor(ps, off, 32);
      ns += __shfl_xor(ns, off, 32);
    }
    if (lane == 0) {
      const float Lk = (float)(Tv - (wid + 1));
      sk[wid] = ps / ((float)Bv * Lk) + ns / ((float)NN * (float)Bv * Lk);
    }
  }
  __syncthreads();
  if (threadIdx.x == 0) {
    float t = 0.f;
    for (int i = 0; i < Kv; ++i) t += sk[i];
    out[0] = t / (float)Kv;
  }
}

// ---------------------------------------------------------------------------
extern "C" void kernel_launch(void* const* d_in, const int* in_sizes, int n_in,
                              void* d_out, int out_size, void* d_ws, size_t ws_size,
                              hipStream_t stream) {
  (void)in_sizes; (void)n_in; (void)out_size; (void)ws_size;
  const float* z = (const float*)d_in[0];
  // d_in[1] = c, unused by the reference computation
  const float* pred = (const float*)d_in[2];
  const int* nidx = (const int*)d_in[3];

  // workspace layout: [0, 64KB) per-block partials; [64KB, 64KB+32MB) zT
  float* accum = (float*)d_ws;
  float* zT = (float*)((char*)d_ws + 65536);

  cpc_transpose_z<<<dim3(Tv / 32, Cv / 32, Bv), dim3(32, 8, 1), 0, stream>>>(z, zT);
  cpc_main<<<dim3(LTILES, Bv, Kv), 256, 0, stream>>>(z, pred, nidx, zT, accum);
  cpc_finalize<<<1, 384, 0, stream>>>(accum, (float*)d_out);
}